// EmbeddingBag_26182120636875
// MI455X (gfx1250) — compile-verified
//
#include <hip/hip_runtime.h>
#include <stdint.h>

// EmbeddingBag on MI455X (gfx1250): pure random-gather, HBM-bandwidth bound
// (~220 MB moved, ~9.4 us floor at 23.3 TB/s). WMMA is useless here (per-bag
// private gathered matrices, trivial FLOPs); instead we use the CDNA5 async
// global->LDS engine (ASYNCcnt) to double-buffer the 256B row gathers.

#define BAG 50
#define DIM 64
#define CHUNK 10
#define NCHUNK (BAG / CHUNK)          // 5
#define WAVES_PER_BLOCK 8
#define BLOCK_THREADS (WAVES_PER_BLOCK * 32)

// One async instruction moves one full 256B embedding row:
// 32 lanes x 8B, GVS addressing: mem = SADDR(values) + VADDR32 + 0.
// idx*256 <= 256e6 < 2^31 so the signed 32-bit VGPR offset is safe.
__device__ __forceinline__ void async_row_load(unsigned lds_byte_addr,
                                               unsigned global_off,
                                               const float* base) {
  asm volatile("global_load_async_to_lds_b64 %0, %1, %2"
               :: "v"(lds_byte_addr), "v"(global_off), "s"(base)
               : "memory");
}

__global__ __launch_bounds__(BLOCK_THREADS) void
embedding_bag_async_kernel(const int* __restrict__ indices,
                           const float* __restrict__ weights,
                           const float* __restrict__ values,
                           float* __restrict__ out,
                           int batch) {
  // 8 waves x double buffer x 10 rows x 256B = 40 KB (WGP has 320 KB).
  __shared__ float smem[WAVES_PER_BLOCK][2][CHUNK][DIM];

  const int lane = threadIdx.x & 31;
  const int wv   = threadIdx.x >> 5;
  const int b    = blockIdx.x * WAVES_PER_BLOCK + wv;   // one wave per bag
  if (b >= batch) return;

  const int*   idxp = indices + (size_t)b * BAG;
  const float* wp   = weights + (size_t)b * BAG;

  const unsigned lane8 = (unsigned)lane * 8u;
  // Generic pointer to LDS: low 32 bits are the workgroup-relative LDS byte
  // address (aperture lives in the high 32 bits) -> valid async VDST operand.
  const unsigned lds_wave = (unsigned)(uintptr_t)&smem[wv][0][0][0];

  // ---- prologue: stage chunk 0 into buffer 0 -------------------------------
#pragma unroll
  for (int r = 0; r < CHUNK; ++r) {
    const unsigned i = (unsigned)idxp[r];
    async_row_load(lds_wave + (unsigned)(r * DIM * 4) + lane8,
                   i * (unsigned)(DIM * 4) + lane8, values);
  }

  float2 acc = make_float2(0.0f, 0.0f);

#pragma unroll
  for (int c = 0; c < NCHUNK; ++c) {
    const int nb = c + 1;
    if (nb < NCHUNK) {
      // Make sure our previous DS reads of this buffer are fully retired
      // before the async engine may overwrite it (WAR guard).
      asm volatile("s_wait_dscnt 0x0" ::: "memory");
      const unsigned buf_off = (unsigned)((nb & 1) * CHUNK * DIM * 4);
#pragma unroll
      for (int r = 0; r < CHUNK; ++r) {
        const unsigned i = (unsigned)idxp[nb * CHUNK + r];
        async_row_load(lds_wave + buf_off + (unsigned)(r * DIM * 4) + lane8,
                       i * (unsigned)(DIM * 4) + lane8, values);
      }
      // 20 async ops outstanding; they complete in order, so <=10 means the
      // 10 rows of chunk c have landed in LDS while chunk c+1 is in flight.
      asm volatile("s_wait_asynccnt 0xa" ::: "memory");
    } else {
      asm volatile("s_wait_asynccnt 0x0" ::: "memory");
    }

    // ---- consume chunk c from LDS: 2 fp32 per lane per row ----------------
    const float2* rows = (const float2*)&smem[wv][c & 1][0][0];
#pragma unroll
    for (int r = 0; r < CHUNK; ++r) {
      const float  wn = wp[c * CHUNK + r];     // uniform across the wave
      const float2 v  = rows[r * (DIM / 2) + lane];   // ds_load_b64
      acc.x = fmaf(wn, v.x, acc.x);
      acc.y = fmaf(wn, v.y, acc.y);
    }
  }

  // Coalesced 256B store per wave (float2 per lane).
  float2* outp = (float2*)(out + (size_t)b * DIM);
  outp[lane] = acc;
}

extern "C" void kernel_launch(void* const* d_in, const int* in_sizes, int n_in,
                              void* d_out, int out_size, void* d_ws, size_t ws_size,
                              hipStream_t stream) {
  const int*   indices = (const int*)d_in[0];
  const float* weights = (const float*)d_in[1];
  const float* values  = (const float*)d_in[2];
  float*       out     = (float*)d_out;

  const int batch  = in_sizes[0] / BAG;
  const int blocks = (batch + WAVES_PER_BLOCK - 1) / WAVES_PER_BLOCK;

  embedding_bag_async_kernel<<<blocks, BLOCK_THREADS, 0, stream>>>(
      indices, weights, values, out, batch);
}